// AtomExposureGNN_20212116095605
// MI455X (gfx1250) — compile-verified
//
#include <hip/hip_runtime.h>

// Problem constants (match reference)
#define GH    128   // hidden channels
#define GDIN  64    // input channels
#define GH2   64    // MLP hidden (H/2)

typedef __attribute__((ext_vector_type(2))) float v2f;
typedef __attribute__((ext_vector_type(8))) float v8f;

// ---------------------------------------------------------------------------
// Degree / normalization
// ---------------------------------------------------------------------------
__global__ void deg_init_kernel(float* __restrict__ deg, int n) {
    int i = blockIdx.x * blockDim.x + threadIdx.x;
    if (i < n) deg[i] = 1.0f;               // self-loop contributes 1
}

__global__ void deg_edge_kernel(const int* __restrict__ dst, float* __restrict__ deg, int e) {
    int i = blockIdx.x * blockDim.x + threadIdx.x;
    if (i < e) unsafeAtomicAdd(&deg[dst[i]], 1.0f);
}

__global__ void deg_rsqrt_kernel(float* __restrict__ deg, int n) {
    int i = blockIdx.x * blockDim.x + threadIdx.x;
    if (i < n) deg[i] = rsqrtf(deg[i]);     // deg >= 1 always (self-loops)
}

// ---------------------------------------------------------------------------
// fp32 WMMA GEMM, LDS-staged B:
//   out[M x Nc] = act(A[M x KT] @ B[KT x Nc] + bias),  Nc = NT*16
// One wave owns 16 rows x full Nc (NT accumulators). B is loaded once per
// block into LDS (row stride Nc+8 so the two half-waves of each ds_load hit
// disjoint 16-bank windows: 2*(Nc+8) == 16 mod 64). Each A float2 load
// (global_load_b64) feeds NT back-to-back V_WMMA_F32_16X16X4_F32 ops.
// A operand layout (16x4 f32): lanes 0-15 rows 0-15 K=k0,k0+1; lanes 16-31
// rows 0-15 K=k0+2,k0+3. B symmetric. C/D: VGPR v, lane l ->
// D[(l>=16)*8 + v][l&15].  M % 16 == 0 (true: 50000 = 3125*16).
// ---------------------------------------------------------------------------
template<int KT, int NT>
__global__ void wmma_gemm_f32_lds_kernel(const float* __restrict__ A,
                                         const float* __restrict__ B,
                                         const float* __restrict__ bias,
                                         float* __restrict__ out,
                                         int M, int relu) {
    constexpr int Nc = NT * 16;
    constexpr int LSTR = Nc + 8;            // padded LDS row stride (floats)
    __shared__ float ldsB[KT * LSTR];

    const int tid  = threadIdx.x;
    const int lane = tid & 31;
    const int wib  = tid >> 5;              // wave index in block
    const int wpb  = blockDim.x >> 5;       // waves per block

    // Cooperative stage of B[KT x Nc] into LDS (coalesced rows per wave)
    for (int r = wib; r < KT; r += wpb)
        for (int c = lane; c < Nc; c += 32)
            ldsB[r * LSTR + c] = B[r * Nc + c];
    __syncthreads();

    const int wave = blockIdx.x * wpb + wib;    // one wave per 16-row stripe
    const int mtiles = M >> 4;
    if (wave >= mtiles) return;             // wave-uniform guard, EXEC stays full

    const int lrow  = lane & 15;
    const int khalf = (lane >> 4) << 1;     // 0 for lanes 0-15, 2 for 16-31
    const float* Arow = A + (size_t)(wave * 16 + lrow) * KT;

    v8f acc[NT];
    #pragma unroll
    for (int t = 0; t < NT; ++t) acc[t] = (v8f){};

    #pragma unroll 4
    for (int k0 = 0; k0 < KT; k0 += 4) {
        v2f a = *(const v2f*)(Arow + k0 + khalf);   // global_load_b64
        const float* b0 = ldsB + (k0 + khalf) * LSTR + lrow;
        const float* b1 = b0 + LSTR;
        #pragma unroll
        for (int t = 0; t < NT; ++t) {
            v2f b;
            b.x = b0[t * 16];
            b.y = b1[t * 16];
            acc[t] = __builtin_amdgcn_wmma_f32_16x16x4_f32(
                /*neg_a=*/false, a, /*neg_b=*/false, b,
                /*c_mod=*/(short)0, acc[t], /*reuse_a=*/false, /*reuse_b=*/false);
        }
    }

    const int rowBase = wave * 16 + ((lane >> 4) << 3);  // +8 for lanes 16-31
    #pragma unroll
    for (int t = 0; t < NT; ++t) {
        const int outCol = t * 16 + lrow;
        const float bv = bias ? bias[outCol] : 0.0f;
        #pragma unroll
        for (int v = 0; v < 8; ++v) {
            float val = acc[t][v] + bv;
            if (relu) val = fmaxf(val, 0.0f);
            out[(size_t)(rowBase + v) * Nc + outCol] = val;
        }
    }
}

// ---------------------------------------------------------------------------
// Scatter: self-loop init  agg[i,:] = hw[i,:] * dis[i]^2
// ---------------------------------------------------------------------------
__global__ void scatter_init_kernel(const float* __restrict__ hw,
                                    const float* __restrict__ dis,
                                    float* __restrict__ agg, int total) {
    int idx = blockIdx.x * blockDim.x + threadIdx.x;
    if (idx >= total) return;
    float d = dis[idx >> 7];                // node = idx / 128
    agg[idx] = hw[idx] * d * d;
}

// ---------------------------------------------------------------------------
// Scatter: one wave per edge; lane handles 4 channels (32*4 = 128).
// agg[dst,:] += hw[src,:] * dis[src]*dis[dst]   via hardware fp32 atomics.
// ---------------------------------------------------------------------------
__global__ void scatter_edge_kernel(const float* __restrict__ hw,
                                    const int* __restrict__ src,
                                    const int* __restrict__ dst,
                                    const float* __restrict__ dis,
                                    float* __restrict__ agg, int e) {
    int gid = blockIdx.x * blockDim.x + threadIdx.x;
    int edge = gid >> 5;
    if (edge >= e) return;
    int part = (gid & 31) << 2;             // channel group of 4
    int s = src[edge];
    int d = dst[edge];
    float nrm = dis[s] * dis[d];
    const float4 hv = *(const float4*)(hw + s * GH + part);
    float* ap = agg + d * GH + part;
    unsafeAtomicAdd(ap + 0, hv.x * nrm);
    unsafeAtomicAdd(ap + 1, hv.y * nrm);
    unsafeAtomicAdd(ap + 2, hv.z * nrm);
    unsafeAtomicAdd(ap + 3, hv.w * nrm);
}

// ---------------------------------------------------------------------------
// Fused: agg = relu((agg + bc - rmean) * rsqrt(rvar+eps) * gamma + beta) [+ h_in]
// ---------------------------------------------------------------------------
__global__ void bn_relu_res_kernel(float* __restrict__ agg,
                                   const float* __restrict__ h_in,
                                   const float* __restrict__ bc,
                                   const float* __restrict__ gamma,
                                   const float* __restrict__ beta,
                                   const float* __restrict__ rmean,
                                   const float* __restrict__ rvar,
                                   int useRes, int total) {
    int idx = blockIdx.x * blockDim.x + threadIdx.x;
    if (idx >= total) return;
    int c = idx & (GH - 1);
    float s = rsqrtf(rvar[c] + 1e-5f) * gamma[c];
    float v = (agg[idx] + bc[c] - rmean[c]) * s + beta[c];
    v = fmaxf(v, 0.0f);
    if (useRes) v += h_in[idx];
    agg[idx] = v;
}

// ---------------------------------------------------------------------------
// Final projection: out[i] = dot(t[i, 0:64], W2) + b2
// ---------------------------------------------------------------------------
__global__ void mlp_out_kernel(const float* __restrict__ t,
                               const float* __restrict__ W2,
                               const float* __restrict__ b2,
                               float* __restrict__ out, int n) {
    int i = blockIdx.x * blockDim.x + threadIdx.x;
    if (i >= n) return;
    const float* tr = t + i * GH2;
    float acc = 0.0f;
    #pragma unroll 4
    for (int j = 0; j < GH2; ++j) acc += tr[j] * W2[j];
    out[i] = acc + b2[0];
}

// ---------------------------------------------------------------------------
// Host-side orchestration
// ---------------------------------------------------------------------------
extern "C" void kernel_launch(void* const* d_in, const int* in_sizes, int n_in,
                              void* d_out, int out_size, void* d_ws, size_t ws_size,
                              hipStream_t stream) {
    const float* x     = (const float*)d_in[0];
    const int*   eidx  = (const int*)d_in[1];
    const float* W_in  = (const float*)d_in[2];
    const float* b_in  = (const float*)d_in[3];
    const float* Wc    = (const float*)d_in[4];
    const float* bc    = (const float*)d_in[5];
    const float* gamma = (const float*)d_in[6];
    const float* beta  = (const float*)d_in[7];
    const float* rmean = (const float*)d_in[8];
    const float* rvar  = (const float*)d_in[9];
    const float* W1    = (const float*)d_in[10];
    const float* b1    = (const float*)d_in[11];
    const float* W2    = (const float*)d_in[12];
    const float* b2    = (const float*)d_in[13];

    const int N = in_sizes[0] / GDIN;       // 50000
    const int E = in_sizes[1] / 2;          // 800000
    const int L = in_sizes[5] / GH;         // 3
    const int*   src = eidx;                // edge_index[0]
    const int*   dst = eidx + E;            // edge_index[1]

    // Workspace layout (floats), 256B-aligned chunks
    const size_t nh = (size_t)N * GH;
    const size_t chunk = (nh * 4 + 255) & ~(size_t)255;
    char* wsb = (char*)d_ws;
    float* buf0 = (float*)(wsb);                 // N*H
    float* buf1 = (float*)(wsb + chunk);         // N*H
    float* buf2 = (float*)(wsb + 2 * chunk);     // N*H
    float* dis  = (float*)(wsb + 3 * chunk);     // N

    const int TPB = 256;
    const int wpb = TPB / 32;

    // 1) degree -> dis
    deg_init_kernel<<<(N + TPB - 1) / TPB, TPB, 0, stream>>>(dis, N);
    deg_edge_kernel<<<(E + TPB - 1) / TPB, TPB, 0, stream>>>(dst, dis, E);
    deg_rsqrt_kernel<<<(N + TPB - 1) / TPB, TPB, 0, stream>>>(dis, N);

    const int mtiles = N / 16;
    const int ggrid  = (mtiles + wpb - 1) / wpb;

    // 2) h = relu(x @ W_in + b_in)   [N x 64] @ [64 x 128]
    float* h   = buf0;
    float* hw  = buf1;
    float* agg = buf2;
    wmma_gemm_f32_lds_kernel<GDIN, GH / 16><<<ggrid, TPB, 0, stream>>>(
        x, W_in, b_in, h, N, 1);

    // 3) GCN layers
    const int totNH = N * GH;
    for (int i = 0; i < L; ++i) {
        // hw = h @ Wc[i]
        wmma_gemm_f32_lds_kernel<GH, GH / 16><<<ggrid, TPB, 0, stream>>>(
            h, Wc + (size_t)i * GH * GH, nullptr, hw, N, 0);
        // agg = self-loop + edge scatter
        scatter_init_kernel<<<(totNH + TPB - 1) / TPB, TPB, 0, stream>>>(hw, dis, agg, totNH);
        {
            long long tot = (long long)E * 32;
            int sgrid = (int)((tot + TPB - 1) / TPB);
            scatter_edge_kernel<<<sgrid, TPB, 0, stream>>>(hw, src, dst, dis, agg, E);
        }
        // agg = relu(bn(agg + bc)) [+ h]
        bn_relu_res_kernel<<<(totNH + TPB - 1) / TPB, TPB, 0, stream>>>(
            agg, h, bc + i * GH, gamma + i * GH, beta + i * GH,
            rmean + i * GH, rvar + i * GH, (i > 0) ? 1 : 0, totNH);
        // swap: new h lives in agg's buffer; old h buffer becomes next agg
        float* tmp = h; h = agg; agg = tmp;
    }

    // 4) t = relu(h @ W1 + b1)   [N x 128] @ [128 x 64]   (reuse hw buffer)
    float* t = hw;
    wmma_gemm_f32_lds_kernel<GH, GH2 / 16><<<ggrid, TPB, 0, stream>>>(
        h, W1, b1, t, N, 1);

    // 5) out = t @ W2 + b2
    mlp_out_kernel<<<(N + TPB - 1) / TPB, TPB, 0, stream>>>(t, W2, b2, (float*)d_out, N);
}